// CropSplit_51874615001701
// MI455X (gfx1250) — compile-verified
//
#include <hip/hip_runtime.h>
#include <cstdint>

// Problem constants (from the reference): data (4, 200, 200, 200) f32,
// rois (200, 4) f32 rows = {x1, y1, x2, y2}, out (200, 200, 200) f32.
#define CGRID 2
#define HDIM 200
#define WDIM 200
#define NPTS 200
#define YPB 8        // y rows processed per block
#define BLK 224      // 7 wave32 waves; lanes 200..223 idle

__global__ __launch_bounds__(BLK) void CropSplit_kernel(
    const float* __restrict__ data,
    const float* __restrict__ rois,
    float* __restrict__ out)
{
    // 200 ROIs x 16 B = 3200 B staged in LDS once per block.
    __shared__ float s_rois[NPTS * 4];

    const int tid = threadIdx.x;

    // --- CDNA5 async global->LDS staging of the ROI table -----------------
    // Each active lane copies its own 16-byte ROI row from the global address rois + goff to LDS address lds_addr.
    // GVS form: saddr = rois base (SGPR pair), vaddr = per-lane 32-bit offset.
    if (tid < NPTS) {
        unsigned lds_addr = (unsigned)(uintptr_t)(&s_rois[0]) + (unsigned)tid * 16u;
        unsigned goff     = (unsigned)tid * 16u;
        asm volatile("global_load_async_to_lds_b128 %0, %1, %2"
                     :
                     : "v"(lds_addr), "v"(goff), "s"(rois)
                     : "memory");
    }
    // Each wave waits for its own async copies, then the block barrier makes
    // every wave's slice visible to all.
    asm volatile("s_wait_asynccnt 0" ::: "memory");
    __syncthreads();

    const int n = tid;
    if (n >= NPTS) return;   // no barriers past this point

    const int x  = blockIdx.x;
    const int y0 = blockIdx.y * YPB;

    // Per-thread ROI parameters (ds_load_b128), hoisted out of the y loop.
    const float4 r  = reinterpret_cast<const float4*>(s_rois)[n];
    const float x1 = r.x, y1 = r.y, x2 = r.z, y2 = r.w;

    const float fx = (float)x;
    const bool insideX = (fx >= x1) && (fx <= x2);
    const float cw  = fmaxf(x2 - x1, 1.0f);
    // clip(floor(2*(fx-x1)/cw), 0, 1) == (q >= 1.0f).  IEEE divide to match
    // the reference's floor boundaries exactly.
    const float qx  = (2.0f * (fx - x1)) / cw;
    const unsigned cx = (qx >= 1.0f) ? 1u : 0u;
    const float chh = fmaxf(y2 - y1, 1.0f);

    const unsigned xn = (unsigned)x * NPTS + (unsigned)n;     // (x, n) offset
    const unsigned planeStride = (unsigned)HDIM * WDIM * NPTS; // 8,000,000

#pragma unroll
    for (int j = 0; j < YPB; ++j) {
        const int y = y0 + j;
        const float fy = (float)y;
        const bool inside = insideX && (fy >= y1) && (fy <= y2);

        const float qy  = (2.0f * (fy - y1)) / chh;
        const unsigned cy = (qy >= 1.0f) ? 1u : 0u;
        const unsigned cell = cy * CGRID + cx;

        // 32-bit indexing: max index 4*8e6 < 2^31 -> saddr + scaled voffset.
        const unsigned rowOff = (unsigned)y * (WDIM * NPTS) + xn;
        const float v = data[cell * planeStride + rowOff];   // gather (<=4 lines/wave)
        out[rowOff] = inside ? v : 0.0f;                     // coalesced 128B/wave
    }
}

extern "C" void kernel_launch(void* const* d_in, const int* in_sizes, int n_in,
                              void* d_out, int out_size, void* d_ws, size_t ws_size,
                              hipStream_t stream) {
    const float* data = (const float*)d_in[0];   // (4, 200, 200, 200) f32
    const float* rois = (const float*)d_in[1];   // (200, 4) f32
    float* out = (float*)d_out;                  // (200, 200, 200) f32

    dim3 grid(WDIM, HDIM / YPB);  // (200, 25) = 5000 blocks
    dim3 block(BLK);
    CropSplit_kernel<<<grid, block, 0, stream>>>(data, rois, out);
}